// NodeBlock_44564580663324
// MI455X (gfx1250) — compile-verified
//
#include <hip/hip_runtime.h>
#include <hip/hip_bf16.h>

#define N_NODES 100000
#define N_EDGES 1600000
#define DIM 64
#define HID 128
#define TILES (N_NODES / 16)   // 6250, exact
#define WAVES_PER_BLOCK 4
#define LN_EPS 1e-5f

typedef __attribute__((ext_vector_type(2))) float v2f;
typedef __attribute__((ext_vector_type(8))) float v8f;

// ---------------------------------------------------------------------------
// Kernel 1: copy efeat -> d_out[0 : E*D)  (tuple output #0), 16B vectorized
// ---------------------------------------------------------------------------
__global__ __launch_bounds__(256) void copy_efeat_kernel(
    const float4* __restrict__ src, float4* __restrict__ dst, int n4) {
  int i = blockIdx.x * 256 + threadIdx.x;
  if (i < n4) dst[i] = src[i];
}

// ---------------------------------------------------------------------------
// Kernel 2: scatter-add edge features into agg[N, 64] with native f32 atomics
// (agg is 25.6 MB -> resident in 192 MB L2; atomics resolve at L2)
// One thread handles a float2 of one edge row.
// ---------------------------------------------------------------------------
__global__ __launch_bounds__(256) void scatter_add_kernel(
    const float* __restrict__ efeat, const int* __restrict__ dst_idx,
    float* __restrict__ agg, int total2) {
  int i = blockIdx.x * 256 + threadIdx.x;
  if (i >= total2) return;
  int e = i >> 5;            // edge id
  int d = (i & 31) * 2;      // feature pair
  int n = dst_idx[e];
  const float* src = efeat + (size_t)e * DIM + d;
  float* p = agg + (size_t)n * DIM + d;
  unsafeAtomicAdd(p, src[0]);
  unsafeAtomicAdd(p + 1, src[1]);
}

// ---------------------------------------------------------------------------
// Kernel 3: per-node MLP + LayerNorm + residual, fp32 WMMA (16x16x4_f32).
// One wave = one 16-node tile. Block = 4 waves.
// ---------------------------------------------------------------------------
__global__ __launch_bounds__(128) void node_mlp_kernel(
    const float* __restrict__ agg,    // [N,64]
    const float* __restrict__ nfeat,  // [N,64]
    const float* __restrict__ W1,     // [128,128] row-major (k, n)
    const float* __restrict__ b1,     // [128]
    const float* __restrict__ W2,     // [128,64]  row-major (k, n)
    const float* __restrict__ b2,     // [64]
    const float* __restrict__ gamma_, // [64]
    const float* __restrict__ beta_,  // [64]
    float* __restrict__ out)          // [N,64]
{
  // per-wave scratch: 16x130 floats for h1 (reused for h2 @ stride 65) + stats
  __shared__ float lds[WAVES_PER_BLOCK * 16 * 130];

  const int lane = threadIdx.x & 31;
  const int wave = threadIdx.x >> 5;
  const int tile = blockIdx.x * WAVES_PER_BLOCK + wave;
  if (tile >= TILES) return;          // wave-uniform exit: EXEC stays all-ones

  float* h1 = &lds[wave * 16 * 130];
  const int mrow = lane & 15;         // A-frag row / B-frag column-in-tile
  const int hi   = lane >> 4;         // 0: K pair (4t,4t+1); 1: (4t+2,4t+3)
  const int m0   = tile * 16;

  const float* xrowA = agg   + (size_t)(m0 + mrow) * DIM;
  const float* xrowN = nfeat + (size_t)(m0 + mrow) * DIM;

  // ---------------- GEMM1: x[16 x 128] @ W1[128 x 128] ----------------
  v8f acc[8];
#pragma unroll
  for (int nt = 0; nt < 8; ++nt)
#pragma unroll
    for (int r = 0; r < 8; ++r) acc[nt][r] = 0.f;

  for (int t = 0; t < 32; ++t) {
    const int c = 4 * t + 2 * hi;     // K pair (c, c+1); uniform side of concat
    const float* row = (c < DIM) ? (xrowA + c) : (xrowN + (c - DIM));
    v2f a;
    a.x = row[0];
    a.y = row[1];
#pragma unroll
    for (int nt = 0; nt < 8; ++nt) {
      const int n = nt * 16 + mrow;
      v2f b;
      b.x = W1[(size_t)c * HID + n];
      b.y = W1[(size_t)(c + 1) * HID + n];
      acc[nt] = __builtin_amdgcn_wmma_f32_16x16x4_f32(
          false, a, false, b, (short)0, acc[nt], false, false);
    }
  }

  // bias + SiLU, stage h1[16 x 128] in LDS (row stride 130)
#pragma unroll
  for (int nt = 0; nt < 8; ++nt) {
    const int n = nt * 16 + mrow;
    const float bias = b1[n];
#pragma unroll
    for (int r = 0; r < 8; ++r) {
      const int m = r + 8 * hi;
      float v = acc[nt][r] + bias;
      h1[m * 130 + n] = v / (1.f + __expf(-v));   // silu
    }
  }

  // ---------------- GEMM2: h1[16 x 128] @ W2[128 x 64] ----------------
  v8f acc2[4];
#pragma unroll
  for (int nt = 0; nt < 4; ++nt)
#pragma unroll
    for (int r = 0; r < 8; ++r) acc2[nt][r] = 0.f;

  for (int t = 0; t < 32; ++t) {
    const int c = 4 * t + 2 * hi;
    v2f a;
    a.x = h1[mrow * 130 + c];
    a.y = h1[mrow * 130 + c + 1];
#pragma unroll
    for (int nt = 0; nt < 4; ++nt) {
      const int n = nt * 16 + mrow;
      v2f b;
      b.x = W2[(size_t)c * DIM + n];
      b.y = W2[(size_t)(c + 1) * DIM + n];
      acc2[nt] = __builtin_amdgcn_wmma_f32_16x16x4_f32(
          false, a, false, b, (short)0, acc2[nt], false, false);
    }
  }

  // bias, stage h2[16 x 64] (row stride 65; h1 data is dead now)
  float* h2 = h1;
#pragma unroll
  for (int nt = 0; nt < 4; ++nt) {
    const int n = nt * 16 + mrow;
    const float bias = b2[n];
#pragma unroll
    for (int r = 0; r < 8; ++r) {
      const int m = r + 8 * hi;
      h2[m * 65 + n] = acc2[nt][r] + bias;
    }
  }

  // ---------------- LayerNorm stats: lanes 0..15, one node each -------
  if (lane < 16) {
    float s = 0.f;
    for (int d = 0; d < DIM; ++d) s += h2[lane * 65 + d];
    const float mu = s * (1.f / DIM);
    float v = 0.f;
    for (int d = 0; d < DIM; ++d) {
      float t2 = h2[lane * 65 + d] - mu;
      v += t2 * t2;
    }
    h1[1040 + lane] = mu;
    h1[1056 + lane] = __frsqrt_rn(v * (1.f / DIM) + LN_EPS);
  }

  // ---------------- normalize + residual, coalesced stores ------------
  const float* nbase = nfeat + (size_t)m0 * DIM;
  float* obase = out + (size_t)m0 * DIM;
#pragma unroll 4
  for (int i = 0; i < 32; ++i) {
    const int idx = i * 32 + lane;    // 0..1023 over 16 nodes x 64 dims
    const int m = idx >> 6;
    const int d = idx & 63;
    const float mu = h1[1040 + m];
    const float rstd = h1[1056 + m];
    const float val =
        (h2[m * 65 + d] - mu) * rstd * gamma_[d] + beta_[d] + nbase[idx];
    obase[idx] = val;
  }
}

// ---------------------------------------------------------------------------
extern "C" void kernel_launch(void* const* d_in, const int* in_sizes, int n_in,
                              void* d_out, int out_size, void* d_ws,
                              size_t ws_size, hipStream_t stream) {
  const float* efeat = (const float*)d_in[0];
  const float* nfeat = (const float*)d_in[1];
  const int* dst_idx = (const int*)d_in[2];
  const float* W1 = (const float*)d_in[3];
  const float* b1 = (const float*)d_in[4];
  const float* W2 = (const float*)d_in[5];
  const float* b2 = (const float*)d_in[6];
  const float* gamma_ = (const float*)d_in[7];
  const float* beta_ = (const float*)d_in[8];

  float* out_efeat = (float*)d_out;
  float* out_nodes = out_efeat + (size_t)N_EDGES * DIM;
  float* agg = (float*)d_ws;  // [N_NODES, 64] scratch, 25.6 MB

  // zero the scatter accumulator (memset nodes are graph-capturable)
  hipMemsetAsync(agg, 0, (size_t)N_NODES * DIM * sizeof(float), stream);

  // tuple output #0: passthrough efeat
  {
    const int n4 = (N_EDGES * DIM) / 4;  // 25.6M float4
    copy_efeat_kernel<<<(n4 + 255) / 256, 256, 0, stream>>>(
        (const float4*)efeat, (float4*)out_efeat, n4);
  }

  // scatter-sum edges -> agg
  {
    const int total2 = N_EDGES * (DIM / 2);  // 51.2M float2 work items
    scatter_add_kernel<<<(total2 + 255) / 256, 256, 0, stream>>>(
        efeat, dst_idx, agg, total2);
  }

  // node MLP + LayerNorm + residual
  {
    const int blocks = (TILES + WAVES_PER_BLOCK - 1) / WAVES_PER_BLOCK;
    node_mlp_kernel<<<blocks, 32 * WAVES_PER_BLOCK, 0, stream>>>(
        agg, nfeat, W1, b1, W2, b2, gamma_, beta_, out_nodes);
  }
}